// Mamba_17566416241347
// MI455X (gfx1250) — compile-verified
//
#include <hip/hip_runtime.h>
#include <hip/hip_bf16.h>
#include <math.h>

// ---------------------------------------------------------------------------
// Mamba forward for MI455X (gfx1250, wave32).
//   * GEMMs on the fp32 matrix pipe: V_WMMA_F32_16X16X4_F32
//   * Global->LDS staging with GLOBAL_LOAD_ASYNC_TO_LDS_B128 (ASYNCcnt),
//     double-buffered; fragments served from LDS (bank-conflict-free pad).
//   * Selective scan: one thread per (batch, channel), 16-state in registers.
// ---------------------------------------------------------------------------

#define D_MODEL 1024
#define D_STATE 16
#define D_CONV  4
#define D_INNER 2048
#define DT_RANK 64
#define BATCH   2
#define SEQLEN  2048
#define BL      (BATCH * SEQLEN)     // 4096 rows

typedef __attribute__((ext_vector_type(2))) float v2f;
typedef __attribute__((ext_vector_type(8))) float v8f;
typedef __attribute__((ext_vector_type(4))) int   v4i;

#if __has_builtin(__builtin_amdgcn_global_load_async_to_lds_b128)
#define HAVE_ASYNC_LDS 1
#else
#define HAVE_ASYNC_LDS 0
#endif

__device__ __forceinline__ void wait_async0() {
#if __has_builtin(__builtin_amdgcn_s_wait_asynccnt)
    __builtin_amdgcn_s_wait_asynccnt(0);
#else
    asm volatile("s_wait_asynccnt 0" ::: "memory");
#endif
}

__device__ __forceinline__ float softplus_f(float v) {
    return (v > 20.0f) ? v : log1pf(expf(v));
}
__device__ __forceinline__ float silu_f(float v) {
    return v / (1.0f + expf(-v));
}

__device__ __forceinline__ v8f wmma_f32(v2f a, v2f b, v8f c) {
    return __builtin_amdgcn_wmma_f32_16x16x4_f32(false, a, false, b, (short)0, c,
                                                 false, false);
}

// ---------------------------------------------------------------------------
// Staged WMMA fp32 GEMM:  C[M,N] = A[M,K] * Bw[N,K]^T  (row-major, ld params)
// Block: 256 thr = 8 waves; block tile 128x128, K-tile 32, double-buffered LDS.
// Wave (mw = w&3, nw = w>>2) owns a 32x64 chunk = 2x4 WMMA accumulators.
// LDS tiles padded to stride 36 floats (36*r mod 64 distinct for r=0..15).
// EPI==1: C = softplus(acc + bias[col]).
// Requires: M%128==0, N%128==0, K%32==0, lda/ldb multiples of 4.
// ---------------------------------------------------------------------------
#define KTILE    32
#define LDSTRIDE 36

#if HAVE_ASYNC_LDS
__device__ __forceinline__ void issue_tile_async(const float* __restrict__ G, int ld,
                                                 int rbase, int k0,
                                                 float* __restrict__ dst, int tid) {
    // 128 rows x 32 floats = 1024 float4s; 4 per thread, coalesced b128s.
#pragma unroll
    for (int p = 0; p < 4; ++p) {
        const int idx = p * 256 + tid;
        const int r   = idx >> 3;
        const int kq  = (idx & 7) << 2;
        const float* src = G + (size_t)(rbase + r) * ld + k0 + kq;
        float* d = dst + r * LDSTRIDE + kq;
        __builtin_amdgcn_global_load_async_to_lds_b128(
            (__attribute__((address_space(1))) v4i*)src,
            (__attribute__((address_space(3))) v4i*)d, 0, 0);
    }
}

template <int EPI>
__global__ void __launch_bounds__(256)
wmma_gemm_staged(const float* __restrict__ A,
                 const float* __restrict__ Bw,
                 const float* __restrict__ bias,
                 float* __restrict__ C,
                 int M, int N, int K,
                 int lda, int ldb, int ldc) {
    __shared__ float ldsA[2][128 * LDSTRIDE];
    __shared__ float ldsB[2][128 * LDSTRIDE];

    const int tid  = threadIdx.x;
    const int lane = tid & 31;
    const int wave = tid >> 5;
    const int half = lane >> 4;
    const int l16  = lane & 15;
    const int mw   = wave & 3;   // 4 wave-rows of 32
    const int nw   = wave >> 2;  // 2 wave-cols of 64

    const int blockRow = blockIdx.x * 128;
    const int blockCol = blockIdx.y * 128;
    const int rowW = mw * 32;             // within block tile
    const int colW = nw * 64;

    v8f acc[2][4];
#pragma unroll
    for (int m = 0; m < 2; ++m)
#pragma unroll
        for (int n = 0; n < 4; ++n)
#pragma unroll
            for (int i = 0; i < 8; ++i) acc[m][n][i] = 0.0f;

    const int nstages = K / KTILE;
    int buf = 0;

    issue_tile_async(A, lda, blockRow, 0, &ldsA[0][0], tid);
    issue_tile_async(Bw, ldb, blockCol, 0, &ldsB[0][0], tid);

    for (int s = 0; s < nstages; ++s) {
        wait_async0();
        __syncthreads();
        if (s + 1 < nstages) {
            issue_tile_async(A, lda, blockRow, (s + 1) * KTILE, &ldsA[buf ^ 1][0], tid);
            issue_tile_async(Bw, ldb, blockCol, (s + 1) * KTILE, &ldsB[buf ^ 1][0], tid);
        }
        const float* lA = &ldsA[buf][0];
        const float* lB = &ldsB[buf][0];
#pragma unroll
        for (int kk = 0; kk < KTILE; kk += 4) {
            v2f aF[2], bF[4];
#pragma unroll
            for (int m = 0; m < 2; ++m)
                aF[m] = *(const v2f*)&lA[(rowW + m * 16 + l16) * LDSTRIDE + kk + 2 * half];
#pragma unroll
            for (int n = 0; n < 4; ++n)
                bF[n] = *(const v2f*)&lB[(colW + n * 16 + l16) * LDSTRIDE + kk + 2 * half];
#pragma unroll
            for (int m = 0; m < 2; ++m)
#pragma unroll
                for (int n = 0; n < 4; ++n)
                    acc[m][n] = wmma_f32(aF[m], bF[n], acc[m][n]);
        }
        buf ^= 1;
    }

#pragma unroll
    for (int m = 0; m < 2; ++m)
#pragma unroll
        for (int n = 0; n < 4; ++n) {
            const int col = blockCol + colW + n * 16 + l16;
            float bval = 0.0f;
            if (EPI == 1) bval = bias[col];
            const int rbase = blockRow + rowW + m * 16 + 8 * half;
#pragma unroll
            for (int r = 0; r < 8; ++r) {
                float v = acc[m][n][r];
                if (EPI == 1) v = softplus_f(v + bval);
                C[(size_t)(rbase + r) * ldc + col] = v;
            }
        }
}
#endif  // HAVE_ASYNC_LDS

// ---------------------------------------------------------------------------
// Direct (non-staged) WMMA GEMM. Used for x_proj (N=96) and as fallback.
// ---------------------------------------------------------------------------
template <int NTILES, int EPI>
__global__ void wmma_gemm_direct(const float* __restrict__ A,
                                 const float* __restrict__ Bw,
                                 const float* __restrict__ bias,
                                 float* __restrict__ C,
                                 int M, int N, int K,
                                 int lda, int ldb, int ldc) {
    const int lane = threadIdx.x & 31;
    const int wave = threadIdx.x >> 5;
    const int half = lane >> 4;
    const int l16  = lane & 15;

    const int row0 = blockIdx.x * 128 + wave * 16;
    const int col0 = blockIdx.y * (16 * NTILES);

    v8f acc[NTILES];
#pragma unroll
    for (int t = 0; t < NTILES; ++t)
#pragma unroll
        for (int i = 0; i < 8; ++i) acc[t][i] = 0.0f;

    const float* Abase = A + (size_t)(row0 + l16) * lda + 2 * half;

    for (int k0 = 0; k0 < K; k0 += 4) {
        v2f afrag;
        afrag.x = Abase[k0 + 0];
        afrag.y = Abase[k0 + 1];
#pragma unroll
        for (int t = 0; t < NTILES; ++t) {
            const float* Bp = Bw + (size_t)(col0 + t * 16 + l16) * ldb + k0 + 2 * half;
            v2f bfrag;
            bfrag.x = Bp[0];
            bfrag.y = Bp[1];
            acc[t] = wmma_f32(afrag, bfrag, acc[t]);
        }
    }

#pragma unroll
    for (int t = 0; t < NTILES; ++t) {
        const int col = col0 + t * 16 + l16;
        float bval = 0.0f;
        if (EPI == 1) bval = bias[col];
#pragma unroll
        for (int r = 0; r < 8; ++r) {
            float v = acc[t][r];
            if (EPI == 1) v = softplus_f(v + bval);
            C[(size_t)(row0 + r + 8 * half) * ldc + col] = v;
        }
    }
}

// ---------------------------------------------------------------------------
// Depthwise causal conv (width 4) + bias + SiLU.
// ---------------------------------------------------------------------------
__global__ void conv_silu_kernel(const float* __restrict__ xz,
                                 const float* __restrict__ conv_w,
                                 const float* __restrict__ conv_b,
                                 float* __restrict__ x_act) {
    int i = blockIdx.x * blockDim.x + threadIdx.x;
    if (i >= BL * D_INNER) return;
    const int d   = i & (D_INNER - 1);
    const int row = i >> 11;
    const int l   = row & (SEQLEN - 1);
    const int b   = row >> 11;

    float acc = conv_b[d];
#pragma unroll
    for (int j = 0; j < D_CONV; ++j) {
        const int ls = l - (D_CONV - 1) + j;
        if (ls >= 0) {
            const size_t src = (size_t)(b * SEQLEN + ls) * (2 * D_INNER) + d;
            acc += xz[src] * conv_w[d * D_CONV + j];
        }
    }
    x_act[(size_t)row * D_INNER + d] = silu_f(acc);
}

// ---------------------------------------------------------------------------
// Selective scan: one thread per (b, d); 16 states in registers; B/C in LDS.
// Fuses D-skip and SiLU(z) gate; y overwrites x_act in place.
// ---------------------------------------------------------------------------
__global__ void scan_kernel(const float* __restrict__ xz,
                            float* __restrict__ x_act,
                            const float* __restrict__ x_dbl,
                            const float* __restrict__ dt_full,
                            const float* __restrict__ A_log,
                            const float* __restrict__ Dvec) {
    __shared__ float sB[D_STATE];
    __shared__ float sC[D_STATE];

    const int b = blockIdx.y;
    const int d = blockIdx.x * blockDim.x + threadIdx.x;

    float Arow[D_STATE];
#pragma unroll
    for (int n = 0; n < D_STATE; ++n)
        Arow[n] = -expf(A_log[(size_t)d * D_STATE + n]);
    const float Dd = Dvec[d];

    float h[D_STATE];
#pragma unroll
    for (int n = 0; n < D_STATE; ++n) h[n] = 0.0f;

    for (int l = 0; l < SEQLEN; ++l) {
        const int row = b * SEQLEN + l;
        if (threadIdx.x < D_STATE) {
            sB[threadIdx.x] = x_dbl[(size_t)row * 96 + DT_RANK + threadIdx.x];
        } else if (threadIdx.x < 2 * D_STATE) {
            const int n = threadIdx.x - D_STATE;
            sC[n] = x_dbl[(size_t)row * 96 + DT_RANK + D_STATE + n];
        }
        __syncthreads();

        const float dt = dt_full[(size_t)row * D_INNER + d];
        const float xv = x_act[(size_t)row * D_INNER + d];
        const float zv = xz[(size_t)row * (2 * D_INNER) + D_INNER + d];

        float y = 0.0f;
        const float dtx = dt * xv;
#pragma unroll
        for (int n = 0; n < D_STATE; ++n) {
            const float dA = expf(dt * Arow[n]);
            h[n] = dA * h[n] + dtx * sB[n];
            y += h[n] * sC[n];
        }
        y = (y + Dd * xv) * silu_f(zv);
        x_act[(size_t)row * D_INNER + d] = y;
        __syncthreads();
    }
}

// ---------------------------------------------------------------------------
// Launcher
// ---------------------------------------------------------------------------
extern "C" void kernel_launch(void* const* d_in, const int* in_sizes, int n_in,
                              void* d_out, int out_size, void* d_ws, size_t ws_size,
                              hipStream_t stream) {
    const float* hidden    = (const float*)d_in[0];  // BL x 1024
    const float* in_proj_w = (const float*)d_in[1];  // 4096 x 1024
    const float* conv_w    = (const float*)d_in[2];  // 2048 x 4
    const float* conv_b    = (const float*)d_in[3];  // 2048
    const float* x_proj_w  = (const float*)d_in[4];  // 96 x 2048
    const float* dt_proj_w = (const float*)d_in[5];  // 2048 x 64
    const float* dt_proj_b = (const float*)d_in[6];  // 2048
    const float* A_log     = (const float*)d_in[7];  // 2048 x 16
    const float* Dvec      = (const float*)d_in[8];  // 2048
    const float* out_projw = (const float*)d_in[9];  // 1024 x 2048
    float* out = (float*)d_out;                      // BL x 1024

    float* ws    = (float*)d_ws;
    float* xz    = ws;                                  // BL x 4096 (x | z)
    float* x_act = xz    + (size_t)BL * 2 * D_INNER;    // BL x 2048 (x, then y)
    float* x_dbl = x_act + (size_t)BL * D_INNER;        // BL x 96
    float* dt_f  = x_dbl + (size_t)BL * 96;             // BL x 2048

#if HAVE_ASYNC_LDS
    // 1) in_proj: xz = hidden @ in_proj_w^T   (M=4096, N=4096, K=1024)
    wmma_gemm_staged<0><<<dim3(BL / 128, (2 * D_INNER) / 128), 256, 0, stream>>>(
        hidden, in_proj_w, nullptr, xz,
        BL, 2 * D_INNER, D_MODEL, D_MODEL, D_MODEL, 2 * D_INNER);
#else
    wmma_gemm_direct<4, 0><<<dim3(BL / 128, (2 * D_INNER) / 64), 256, 0, stream>>>(
        hidden, in_proj_w, nullptr, xz,
        BL, 2 * D_INNER, D_MODEL, D_MODEL, D_MODEL, 2 * D_INNER);
#endif

    // 2) depthwise conv + SiLU -> x_act
    conv_silu_kernel<<<(BL * D_INNER) / 256, 256, 0, stream>>>(
        xz, conv_w, conv_b, x_act);

    // 3) x_proj: x_dbl = x_act @ x_proj_w^T   (M=4096, N=96, K=2048)
    wmma_gemm_direct<1, 0><<<dim3(BL / 128, 96 / 16), 256, 0, stream>>>(
        x_act, x_proj_w, nullptr, x_dbl,
        BL, 96, D_INNER, D_INNER, D_INNER, 96);

    // 4) dt_proj + softplus: dt_f = softplus(x_dbl[:, :64] @ dt_proj_w^T + b)
#if HAVE_ASYNC_LDS
    wmma_gemm_staged<1><<<dim3(BL / 128, D_INNER / 128), 256, 0, stream>>>(
        x_dbl, dt_proj_w, dt_proj_b, dt_f,
        BL, D_INNER, DT_RANK, 96, DT_RANK, D_INNER);
#else
    wmma_gemm_direct<4, 1><<<dim3(BL / 128, D_INNER / 64), 256, 0, stream>>>(
        x_dbl, dt_proj_w, dt_proj_b, dt_f,
        BL, D_INNER, DT_RANK, 96, DT_RANK, D_INNER);
#endif

    // 5) selective scan (y overwrites x_act)
    scan_kernel<<<dim3(D_INNER / 256, BATCH), 256, 0, stream>>>(
        xz, x_act, x_dbl, dt_f, A_log, Dvec);

    // 6) out_proj: out = y @ out_proj_w^T   (M=4096, N=1024, K=2048)
#if HAVE_ASYNC_LDS
    wmma_gemm_staged<0><<<dim3(BL / 128, D_MODEL / 128), 256, 0, stream>>>(
        x_act, out_projw, nullptr, out,
        BL, D_MODEL, D_INNER, D_INNER, D_INNER, D_MODEL);
#else
    wmma_gemm_direct<4, 0><<<dim3(BL / 128, D_MODEL / 64), 256, 0, stream>>>(
        x_act, out_projw, nullptr, out,
        BL, D_MODEL, D_INNER, D_INNER, D_INNER, D_MODEL);
#endif
}